// GraphSage_2671469658134
// MI455X (gfx1250) — compile-verified
//
#include <hip/hip_runtime.h>
#include <math.h>

#define FDIM 128  // feature dim D, compile-time specialized

typedef __attribute__((ext_vector_type(2))) float v2f;
typedef __attribute__((ext_vector_type(8))) float v8f;

__device__ __forceinline__ void atomAddF(float* p, float v) {
    (void)__hip_atomic_fetch_add(p, v, __ATOMIC_RELAXED, __HIP_MEMORY_SCOPE_AGENT);
}

// --- degree count: one thread per edge -------------------------------------
__global__ void sage_deg_kernel(const int* __restrict__ dst, float* __restrict__ deg, int E) {
    int e = blockIdx.x * blockDim.x + threadIdx.x;
    if (e < E) atomAddF(&deg[dst[e]], 1.0f);
}

// --- invert degrees in place: inv = 1/max(deg,1) ---------------------------
__global__ void sage_invdeg_kernel(float* __restrict__ deg, int n) {
    int i = blockIdx.x * blockDim.x + threadIdx.x;
    if (i < n) deg[i] = 1.0f / fmaxf(deg[i], 1.0f);
}

// --- edge gather + scatter-add: one wave per edge, lane owns 4 features ----
__global__ __launch_bounds__(256)
void sage_edge_kernel(const float* __restrict__ x, const int* __restrict__ src,
                      const int* __restrict__ dst, const float* __restrict__ ew,
                      float* __restrict__ agg, int E) {
    long gid = (long)blockIdx.x * blockDim.x + threadIdx.x;
    int e = (int)(gid >> 5);
    if (e >= E) return;
    int t = (int)(gid & 31);                 // feature group: 4 floats
    int s = src[e];
    int d = dst[e];
    float w = ew[e];
    const float4 v = *((const float4*)(x + (size_t)s * FDIM) + t);
    float* ap = agg + (size_t)d * FDIM + t * 4;
    atomAddF(ap + 0, w * v.x);
    atomAddF(ap + 1, w * v.y);
    atomAddF(ap + 2, w * v.z);
    atomAddF(ap + 3, w * v.w);
}

// --- fused dual-GEMM + bias + activation via V_WMMA_F32_16X16X4_F32 --------
// Block = 256 threads = 8 waves. Block owns 16 node rows; wave w owns output
// column tile [16*w, 16*w+16). Accumulates (inv_deg*agg)@Wn + x@Ws over K=128.
__global__ __launch_bounds__(256)
void sage_gemm_kernel(const float* __restrict__ agg, const float* __restrict__ xin,
                      const float* __restrict__ invdeg,
                      const float* __restrict__ Wn, const float* __restrict__ Ws,
                      const float* __restrict__ bias, float* __restrict__ out,
                      int nNodes, int act /*0=relu, 1=sigmoid*/) {
    const int lane = threadIdx.x & 31;
    const int wv   = threadIdx.x >> 5;      // column tile 0..7
    const int half = lane >> 4;             // K-half select
    const int mi   = lane & 15;             // row index (A) / col index (B)
    const int m_tile = blockIdx.x * 16;

    int row = m_tile + mi;
    int rowc = row < nNodes ? row : nNodes - 1;   // clamp loads, EXEC stays full
    const int col = wv * 16 + mi;

    const float scale = invdeg[rowc];
    const float* __restrict__ arow = agg + (size_t)rowc * FDIM;
    const float* __restrict__ xrow = xin + (size_t)rowc * FDIM;
    const int kbase = half * 2;

    v8f c = {0.f, 0.f, 0.f, 0.f, 0.f, 0.f, 0.f, 0.f};

    #pragma unroll
    for (int k0 = 0; k0 < FDIM; k0 += 4) {
        const int k = k0 + kbase;
        // A: 32-bit 16x4 layout — lane holds row mi, VGPR v holds K = kbase+v
        v2f a1; a1.x = arow[k] * scale; a1.y = arow[k + 1] * scale;
        // B: 32-bit 4x16 layout — lane holds col mi, VGPR v holds K = kbase+v
        v2f b1; b1.x = Wn[(size_t)k * FDIM + col]; b1.y = Wn[(size_t)(k + 1) * FDIM + col];
        c = __builtin_amdgcn_wmma_f32_16x16x4_f32(false, a1, false, b1, (short)0, c, false, false);

        v2f a2; a2.x = xrow[k]; a2.y = xrow[k + 1];
        v2f b2; b2.x = Ws[(size_t)k * FDIM + col]; b2.y = Ws[(size_t)(k + 1) * FDIM + col];
        c = __builtin_amdgcn_wmma_f32_16x16x4_f32(false, a2, false, b2, (short)0, c, false, false);
    }

    const float bv = bias[col];
    #pragma unroll
    for (int r = 0; r < 8; ++r) {
        int m = m_tile + r + half * 8;      // D layout: VGPR r -> row r + 8*half
        float v = c[r] + bv;
        v = (act == 0) ? fmaxf(v, 0.0f) : (1.0f / (1.0f + __expf(-v)));
        if (m < nNodes) out[(size_t)m * FDIM + col] = v;
    }
}

extern "C" void kernel_launch(void* const* d_in, const int* in_sizes, int n_in,
                              void* d_out, int out_size, void* d_ws, size_t ws_size,
                              hipStream_t stream) {
    const float* x       = (const float*)d_in[0];
    const int*   eidx    = (const int*)d_in[1];
    const float* ew      = (const float*)d_in[2];
    const float* W_self  = (const float*)d_in[3];
    const float* W_neigh = (const float*)d_in[4];
    const float* bias    = (const float*)d_in[5];

    const int nNodes = in_sizes[0] / FDIM;
    const int E      = in_sizes[2];
    const int L      = in_sizes[5] / FDIM;
    float* out = (float*)d_out;

    // workspace layout: [deg | agg | xbufA | xbufB]
    char* w = (char*)d_ws;
    size_t degBytes = (((size_t)nNodes * 4) + 511) & ~(size_t)511;
    float* deg   = (float*)w;
    float* agg   = (float*)(w + degBytes);
    float* xbufA = agg   + (size_t)nNodes * FDIM;
    float* xbufB = xbufA + (size_t)nNodes * FDIM;

    const int* src = eidx;
    const int* dstI = eidx + E;

    // degrees (layer-invariant)
    hipMemsetAsync(deg, 0, (size_t)nNodes * sizeof(float), stream);
    sage_deg_kernel<<<(E + 255) / 256, 256, 0, stream>>>(dstI, deg, E);
    sage_invdeg_kernel<<<(nNodes + 255) / 256, 256, 0, stream>>>(deg, nNodes);

    const float* xcur = x;
    const long edgeThreads = (long)E * 32;
    const int edgeBlocks = (int)((edgeThreads + 255) / 256);
    const int gemmBlocks = (nNodes + 15) / 16;

    for (int l = 0; l < L; ++l) {
        float* y = (l == L - 1) ? out : ((l & 1) ? xbufB : xbufA);

        hipMemsetAsync(agg, 0, (size_t)nNodes * FDIM * sizeof(float), stream);
        sage_edge_kernel<<<edgeBlocks, 256, 0, stream>>>(xcur, src, dstI, ew, agg, E);
        sage_gemm_kernel<<<gemmBlocks, 256, 0, stream>>>(
            agg, xcur, deg,
            W_neigh + (size_t)l * FDIM * FDIM,
            W_self  + (size_t)l * FDIM * FDIM,
            bias    + (size_t)l * FDIM,
            y, nNodes, (l < L - 1) ? 0 : 1);

        xcur = y;
    }
}